// SimpleRNN_9483287789606
// MI455X (gfx1250) — compile-verified
//
#include <hip/hip_runtime.h>

typedef float v2f __attribute__((ext_vector_type(2)));
typedef float v8f __attribute__((ext_vector_type(8)));

#define B_  256
#define T_  512
#define E_  100
#define U_  512
#define D1_ 1024
#define D2_ 1024
#define C_  10

#define NBLK 16          // persistent grid: 16 blocks x 1024 threads = 512 waves

// ---------------------------------------------------------------------------
// Device-wide barrier for the persistent scan kernel.  16 co-resident blocks,
// one L2 atomic arrival per block, generation-flag release, s_sleep spin.
// ---------------------------------------------------------------------------
__device__ __forceinline__ void grid_sync(unsigned* cnt, unsigned* gen) {
  __syncthreads();
  if (threadIdx.x == 0) {
    __threadfence();  // make this block's Hout writes visible device-wide
    unsigned g = __hip_atomic_load(gen, __ATOMIC_ACQUIRE, __HIP_MEMORY_SCOPE_AGENT);
    unsigned t = __hip_atomic_fetch_add(cnt, 1u, __ATOMIC_ACQ_REL, __HIP_MEMORY_SCOPE_AGENT);
    if (t == NBLK - 1u) {
      __hip_atomic_store(cnt, 0u, __ATOMIC_RELEASE, __HIP_MEMORY_SCOPE_AGENT);
      __hip_atomic_fetch_add(gen, 1u, __ATOMIC_ACQ_REL, __HIP_MEMORY_SCOPE_AGENT);
    } else {
      while (__hip_atomic_load(gen, __ATOMIC_ACQUIRE, __HIP_MEMORY_SCOPE_AGENT) == g)
        __builtin_amdgcn_s_sleep(1);
    }
  }
  __syncthreads();
}

// ---------------------------------------------------------------------------
// Persistent RNN scan: all T=512 timesteps in one launch.
//   H[t+1] = mask ? relu(emb[tok_t]@Wx + b_rnn + H[t]@Wh) : H[t]
// 512 waves, one 16x16 f32 WMMA tile per wave per step, grid barrier between
// steps.  H double-buffered in workspace (H0 pre-zeroed; final state in H0
// since T is even).
// ---------------------------------------------------------------------------
__global__ __launch_bounds__(1024) void rnn_scan_kernel(
    const int* __restrict__ tokens, const float* __restrict__ emb,
    const float* __restrict__ Wx, const float* __restrict__ Wh,
    const float* __restrict__ b_rnn,
    float* __restrict__ H0, float* __restrict__ H1,
    unsigned* __restrict__ cnt, unsigned* __restrict__ gen)
{
  const int lane = threadIdx.x & 31;
  const int wave = threadIdx.x >> 5;               // 0..31
  const int tile = blockIdx.x * 32 + wave;         // 0..511
  const int mt = tile & 15;                        // 16 row tiles (B/16)
  const int nt = tile >> 4;                        // 32 col tiles (U/16)
  const int m0 = mt << 4, n0 = nt << 4;
  const int half = lane >> 4;                      // selects K pair
  const int l16  = lane & 15;

  const int rowA = m0 + l16;
  const int col  = n0 + l16;
  const float bn = b_rnn[col];
  const int* tokrow = tokens + rowA * T_;

  float* Hin  = H0;
  float* Hout = H1;

  for (int t = 0; t < T_; ++t) {
    v8f acc = {};

    // ---- fused embedding gather + x_t @ Wx  (K = E = 100 = 25 * 4) ----
    const int tok = tokrow[t];
    const float* arow = emb + tok * E_;
    #pragma unroll 5
    for (int k = 0; k < E_; k += 4) {
      v2f a, b;
      a.x = arow[k + 2 * half];
      a.y = arow[k + 2 * half + 1];
      b.x = Wx[(k + 2 * half)     * U_ + col];
      b.y = Wx[(k + 2 * half + 1) * U_ + col];
      acc = __builtin_amdgcn_wmma_f32_16x16x4_f32(false, a, false, b,
                                                  (short)0, acc, false, false);
    }

    // ---- H @ Wh  (K = U = 512 = 128 * 4) ----
    const float* hrow = Hin + rowA * U_;
    #pragma unroll 4
    for (int k = 0; k < U_; k += 4) {
      v2f a, b;
      a.x = hrow[k + 2 * half];
      a.y = hrow[k + 2 * half + 1];
      b.x = Wh[(k + 2 * half)     * U_ + col];
      b.y = Wh[(k + 2 * half + 1) * U_ + col];
      acc = __builtin_amdgcn_wmma_f32_16x16x4_f32(false, a, false, b,
                                                  (short)0, acc, false, false);
    }

    // ---- bias + relu + masked state carry (branchless select) ----
    #pragma unroll
    for (int i = 0; i < 8; ++i) {
      const int row = m0 + i + 8 * half;           // C/D layout: VGPR i, lane-half
      const int tk  = tokens[row * T_ + t];
      float hnew = acc[i] + bn;
      hnew = hnew > 0.f ? hnew : 0.f;
      const float hold = Hin[row * U_ + col];
      Hout[row * U_ + col] = (tk != 0) ? hnew : hold;
    }

    // all blocks finish step t before anyone starts step t+1
    grid_sync(cnt, gen);

    float* tmp = Hin; Hin = Hout; Hout = tmp;
  }
}

// ---------------------------------------------------------------------------
// Generic C = relu(A[MxK] @ B[KxN] + bias[N]) for the MLP head.
// One 16x16 tile per wave, 8 waves per block.  K % 4 == 0, M,N % 16 == 0.
// ---------------------------------------------------------------------------
__global__ __launch_bounds__(256) void gemm_bias_relu_kernel(
    const float* __restrict__ A, const float* __restrict__ Bm,
    const float* __restrict__ bias, float* __restrict__ Cm,
    int M, int K, int N)
{
  const int lane = threadIdx.x & 31;
  const int wave = threadIdx.x >> 5;
  const int ntn  = N >> 4;
  const int tile = blockIdx.x * 8 + wave;
  const int mt = tile / ntn;
  const int nt = tile % ntn;
  const int m0 = mt << 4, n0 = nt << 4;
  const int half = lane >> 4;
  const int l16  = lane & 15;

  v8f acc = {};
  const float* arow = A + (m0 + l16) * K;
  #pragma unroll 4
  for (int k = 0; k < K; k += 4) {
    v2f a, b;
    a.x = arow[k + 2 * half];
    a.y = arow[k + 2 * half + 1];
    b.x = Bm[(k + 2 * half)     * N + n0 + l16];
    b.y = Bm[(k + 2 * half + 1) * N + n0 + l16];
    acc = __builtin_amdgcn_wmma_f32_16x16x4_f32(false, a, false, b,
                                                (short)0, acc, false, false);
  }

  const int col = n0 + l16;
  const float bn = bias[col];
  #pragma unroll
  for (int i = 0; i < 8; ++i) {
    const int row = m0 + i + 8 * half;
    float v = acc[i] + bn;
    Cm[row * N + col] = v > 0.f ? v : 0.f;
  }
}

// ---------------------------------------------------------------------------
// out[b, :] = softmax(h2[b, :] @ Wo + bo)   (C = 10, K = 1024)
// One wave per batch row; lanes split K, xor-shuffle all-reduce.
// ---------------------------------------------------------------------------
__global__ __launch_bounds__(32) void head_softmax_kernel(
    const float* __restrict__ H, const float* __restrict__ Wo,
    const float* __restrict__ bo, float* __restrict__ out)
{
  const int b = blockIdx.x;
  const int lane = threadIdx.x;
  float acc[C_];
  #pragma unroll
  for (int c = 0; c < C_; ++c) acc[c] = 0.f;

  const float* hrow = H + b * D2_;
  for (int k = lane; k < D2_; k += 32) {
    const float hv = hrow[k];
    const float* wrow = Wo + k * C_;
    #pragma unroll
    for (int c = 0; c < C_; ++c) acc[c] += hv * wrow[c];
  }
  #pragma unroll
  for (int c = 0; c < C_; ++c)
    for (int off = 16; off > 0; off >>= 1)
      acc[c] += __shfl_xor(acc[c], off, 32);

  // every lane now holds all 10 complete dot products
  float logit[C_];
  float m = -1e30f;
  #pragma unroll
  for (int c = 0; c < C_; ++c) { logit[c] = acc[c] + bo[c]; m = fmaxf(m, logit[c]); }
  float s = 0.f;
  #pragma unroll
  for (int c = 0; c < C_; ++c) { logit[c] = expf(logit[c] - m); s += logit[c]; }
  if (lane < C_) out[b * C_ + lane] = logit[lane] / s;
}

// zero H0 and the two barrier words (re-run every call => deterministic)
__global__ void init_kernel(float* __restrict__ h, int n, unsigned* __restrict__ bar) {
  int i = blockIdx.x * blockDim.x + threadIdx.x;
  if (i < n) h[i] = 0.f;
  if (i < 2) bar[i] = 0u;
}

// ---------------------------------------------------------------------------
extern "C" void kernel_launch(void* const* d_in, const int* in_sizes, int n_in,
                              void* d_out, int out_size, void* d_ws, size_t ws_size,
                              hipStream_t stream) {
  const int*   tokens = (const int*)  d_in[0];
  const float* emb    = (const float*)d_in[1];
  const float* Wx     = (const float*)d_in[2];
  const float* Wh     = (const float*)d_in[3];
  const float* b_rnn  = (const float*)d_in[4];
  const float* W1     = (const float*)d_in[5];
  const float* b1     = (const float*)d_in[6];
  const float* W2     = (const float*)d_in[7];
  const float* b2     = (const float*)d_in[8];
  const float* Wo     = (const float*)d_in[9];
  const float* bo     = (const float*)d_in[10];
  float* out = (float*)d_out;

  // workspace: double-buffered hidden state + MLP activations + barrier words
  float* H0 = (float*)d_ws;
  float* H1 = H0 + B_ * U_;
  float* h1 = H1 + B_ * U_;
  float* h2 = h1 + B_ * D1_;
  unsigned* bar = (unsigned*)(h2 + B_ * D2_);   // [0]=cnt, [1]=gen

  init_kernel<<<(B_ * U_ + 255) / 256, 256, 0, stream>>>(H0, B_ * U_, bar);

  // whole 512-step scan in one persistent launch (T even => final state in H0)
  rnn_scan_kernel<<<NBLK, 1024, 0, stream>>>(tokens, emb, Wx, Wh, b_rnn,
                                             H0, H1, bar, bar + 1);

  gemm_bias_relu_kernel<<<(B_ / 16) * (D1_ / 16) / 8, 256, 0, stream>>>(
      H0, W1, b1, h1, B_, U_, D1_);
  gemm_bias_relu_kernel<<<(B_ / 16) * (D2_ / 16) / 8, 256, 0, stream>>>(
      h1, W2, b2, h2, B_, D1_, D2_);
  head_softmax_kernel<<<B_, 32, 0, stream>>>(h2, Wo, bo, out);
}